// MPMSimulator_24094766531209
// MI455X (gfx1250) — compile-verified
//
#include <hip/hip_runtime.h>
#include <stdint.h>

#define GRID 100
#define G3   1000000
#define CHUNK 4096

// ---- simulation constants (match reference exactly) ----
constexpr float DT_       = 5e-4f;
constexpr float INV_DX_   = 100.0f;
constexpr float DX_       = 0.01f;
constexpr float MU_       = 38461.53846153846f;   // 1e5 / 2.6
constexpr float LAM_      = 57692.30769230769f;   // 3e4 / 0.52
constexpr float P_MASS_   = 3.75e-4f;             // 0.005^3 * 3000
constexpr float STRESS_K  = -2.5e-6f;             // -DT * P_VOL * 4 * INV_DX^2
constexpr float C_COEF_   = 40000.0f;             // 4 * INV_DX^2
constexpr int   BOUND_    = 3;

typedef unsigned int v4u  __attribute__((ext_vector_type(4)));
typedef int          v8i_ __attribute__((ext_vector_type(8)));
typedef int          v4i_ __attribute__((ext_vector_type(4)));
typedef float        v2f  __attribute__((ext_vector_type(2)));
typedef float        v8f  __attribute__((ext_vector_type(8)));

__device__ __forceinline__ void atomAddF(float* p, float v) {
    // relaxed, device-scope fp32 atomic -> native global_atomic_add_f32
    __hip_atomic_fetch_add(p, v, __ATOMIC_RELAXED, __HIP_MEMORY_SCOPE_AGENT);
}

// ---------------------------------------------------------------- init
__global__ void init_particles(const float* __restrict__ q, const float* __restrict__ qd,
                               float* __restrict__ px, float* __restrict__ pv,
                               float* __restrict__ pC, float* __restrict__ pF, int N) {
    int p = blockIdx.x * blockDim.x + threadIdx.x;
    if (p >= N) return;
    #pragma unroll
    for (int d = 0; d < 3; ++d) { px[3*p+d] = q[3*p+d]; pv[3*p+d] = qd[3*p+d]; }
    #pragma unroll
    for (int i = 0; i < 9; ++i) { pC[9*p+i] = 0.0f; pF[9*p+i] = (i % 4 == 0) ? 1.0f : 0.0f; }
}

// ---------------------------------------------------------------- grid clear (16 MB, L2-resident)
__global__ void clear_grid(float4* __restrict__ g) {
    int i = blockIdx.x * blockDim.x + threadIdx.x;
    if (i < G3) { float4 z; z.x = z.y = z.z = z.w = 0.0f; g[i] = z; }
}

// ---------------------------------------------------------------- P2G
__global__ void __launch_bounds__(128)
p2g_kernel(const float* __restrict__ px, const float* __restrict__ pv,
           const float* __restrict__ pC, float* __restrict__ pF,
           float* __restrict__ gvx, float* __restrict__ gvy, float* __restrict__ gvz,
           float* __restrict__ gm, int N) {
    int p = blockIdx.x * blockDim.x + threadIdx.x;
    if (p >= N) return;

    float X0 = px[3*p], X1 = px[3*p+1], X2 = px[3*p+2];
    float C[9], F0[9];
    #pragma unroll
    for (int i = 0; i < 9; ++i) C[i]  = pC[9*p+i];
    #pragma unroll
    for (int i = 0; i < 9; ++i) F0[i] = pF[9*p+i];

    // F = (I + dt*C) * F
    float A[9];
    #pragma unroll
    for (int i = 0; i < 9; ++i) A[i] = DT_ * C[i];
    A[0] += 1.0f; A[4] += 1.0f; A[8] += 1.0f;
    float F[9];
    #pragma unroll
    for (int i = 0; i < 3; ++i)
        #pragma unroll
        for (int j = 0; j < 3; ++j)
            F[3*i+j] = A[3*i] * F0[j] + A[3*i+1] * F0[3+j] + A[3*i+2] * F0[6+j];
    #pragma unroll
    for (int i = 0; i < 9; ++i) pF[9*p+i] = F[i];

    // cofactor matrix (invT = cof / det), det, logJ
    float cof[9];
    cof[0] =  F[4]*F[8] - F[5]*F[7];
    cof[1] = -(F[3]*F[8] - F[5]*F[6]);
    cof[2] =  F[3]*F[7] - F[4]*F[6];
    cof[3] = -(F[1]*F[8] - F[2]*F[7]);
    cof[4] =  F[0]*F[8] - F[2]*F[6];
    cof[5] = -(F[0]*F[7] - F[1]*F[6]);
    cof[6] =  F[1]*F[5] - F[2]*F[4];
    cof[7] = -(F[0]*F[5] - F[2]*F[3]);
    cof[8] =  F[0]*F[4] - F[1]*F[3];
    float det  = F[0]*cof[0] + F[1]*cof[1] + F[2]*cof[2];
    float rdet = 1.0f / det;
    float logJ = logf(fmaxf(det, 1e-6f));

    // P = MU*F + (LAM*logJ - MU)*FinvT ; affine = K * P*F^T + P_MASS*C
    float s = LAM_ * logJ - MU_;
    float P[9];
    #pragma unroll
    for (int i = 0; i < 9; ++i) P[i] = MU_ * F[i] + s * (cof[i] * rdet);
    float aff[9];
    #pragma unroll
    for (int i = 0; i < 3; ++i)
        #pragma unroll
        for (int j = 0; j < 3; ++j)
            aff[3*i+j] = STRESS_K * (P[3*i]*F[3*j] + P[3*i+1]*F[3*j+1] + P[3*i+2]*F[3*j+2])
                         + P_MASS_ * C[3*i+j];

    // quadratic B-spline weights
    int base[3]; float fx[3]; float w[3][3];
    float Xv[3] = {X0, X1, X2};
    #pragma unroll
    for (int d = 0; d < 3; ++d) {
        float xs = Xv[d] * INV_DX_;
        float bf = floorf(xs - 0.5f);
        base[d] = (int)bf;
        fx[d] = xs - bf;
        float a0 = 1.5f - fx[d], a1 = fx[d] - 1.0f, a2 = fx[d] - 0.5f;
        w[0][d] = 0.5f * a0 * a0;
        w[1][d] = 0.75f - a1 * a1;
        w[2][d] = 0.5f * a2 * a2;
    }

    float mv0 = P_MASS_ * pv[3*p];
    float mv1 = P_MASS_ * (pv[3*p+1] - DT_ * 9.8f);   // gravity folded into momentum
    float mv2 = P_MASS_ * pv[3*p+2];

    #pragma unroll
    for (int i = 0; i < 3; ++i) {
        float dp0 = ((float)i - fx[0]) * DX_;
        int ni = min(max(base[0] + i, 0), GRID - 1);
        #pragma unroll
        for (int j = 0; j < 3; ++j) {
            float dp1 = ((float)j - fx[1]) * DX_;
            int nj = min(max(base[1] + j, 0), GRID - 1);
            float wij = w[i][0] * w[j][1];
            #pragma unroll
            for (int k = 0; k < 3; ++k) {
                float dp2 = ((float)k - fx[2]) * DX_;
                int nk = min(max(base[2] + k, 0), GRID - 1);
                float wt  = wij * w[k][2];
                int idx = (ni * GRID + nj) * GRID + nk;
                atomAddF(&gvx[idx], wt * (mv0 + aff[0]*dp0 + aff[1]*dp1 + aff[2]*dp2));
                atomAddF(&gvy[idx], wt * (mv1 + aff[3]*dp0 + aff[4]*dp1 + aff[5]*dp2));
                atomAddF(&gvz[idx], wt * (mv2 + aff[6]*dp0 + aff[7]*dp1 + aff[8]*dp2));
                atomAddF(&gm[idx],  wt * P_MASS_);
            }
        }
    }
}

// ---------------------------------------------------------------- G2P (grid normalize + BC fused in)
__global__ void __launch_bounds__(128)
g2p_kernel(float* __restrict__ px, float* __restrict__ pv, float* __restrict__ pC,
           const float* __restrict__ gvx, const float* __restrict__ gvy,
           const float* __restrict__ gvz, const float* __restrict__ gm,
           const float* __restrict__ vptr, int N) {
    int p = blockIdx.x * blockDim.x + threadIdx.x;
    if (p >= N) return;
    float vsc = vptr[0];

    float Xv[3] = {px[3*p], px[3*p+1], px[3*p+2]};
    int base[3]; float fx[3]; float w[3][3];
    #pragma unroll
    for (int d = 0; d < 3; ++d) {
        float xs = Xv[d] * INV_DX_;
        float bf = floorf(xs - 0.5f);
        base[d] = (int)bf;
        fx[d] = xs - bf;
        float a0 = 1.5f - fx[d], a1 = fx[d] - 1.0f, a2 = fx[d] - 0.5f;
        w[0][d] = 0.5f * a0 * a0;
        w[1][d] = 0.75f - a1 * a1;
        w[2][d] = 0.5f * a2 * a2;
    }

    float nv0 = 0.f, nv1 = 0.f, nv2 = 0.f;
    float Cn[9] = {0.f,0.f,0.f,0.f,0.f,0.f,0.f,0.f,0.f};

    #pragma unroll
    for (int i = 0; i < 3; ++i) {
        float dp0 = ((float)i - fx[0]) * DX_;
        int ni = min(max(base[0] + i, 0), GRID - 1);
        #pragma unroll
        for (int j = 0; j < 3; ++j) {
            float dp1 = ((float)j - fx[1]) * DX_;
            int nj = min(max(base[1] + j, 0), GRID - 1);
            float wij = w[i][0] * w[j][1];
            #pragma unroll
            for (int k = 0; k < 3; ++k) {
                float dp2 = ((float)k - fx[2]) * DX_;
                int nk = min(max(base[2] + k, 0), GRID - 1);
                float wt  = wij * w[k][2];
                int idx = (ni * GRID + nj) * GRID + nk;
                // fused grid op: normalize by (mass + v), sticky boundary
                float inv = 1.0f / (gm[idx] + vsc);
                float g0 = gvx[idx] * inv;
                float g1 = gvy[idx] * inv;
                float g2 = gvz[idx] * inv;
                if (ni < BOUND_         && g0 < 0.f) g0 = 0.f;
                if (ni >= GRID - BOUND_ && g0 > 0.f) g0 = 0.f;
                if (nj < BOUND_         && g1 < 0.f) g1 = 0.f;
                if (nj >= GRID - BOUND_ && g1 > 0.f) g1 = 0.f;
                if (nk < BOUND_         && g2 < 0.f) g2 = 0.f;
                if (nk >= GRID - BOUND_ && g2 > 0.f) g2 = 0.f;
                nv0 += wt * g0; nv1 += wt * g1; nv2 += wt * g2;
                float wdp0 = wt * dp0, wdp1 = wt * dp1, wdp2 = wt * dp2;
                Cn[0] += g0 * wdp0; Cn[1] += g0 * wdp1; Cn[2] += g0 * wdp2;
                Cn[3] += g1 * wdp0; Cn[4] += g1 * wdp1; Cn[5] += g1 * wdp2;
                Cn[6] += g2 * wdp0; Cn[7] += g2 * wdp1; Cn[8] += g2 * wdp2;
            }
        }
    }

    px[3*p]   = Xv[0] + DT_ * nv0;
    px[3*p+1] = Xv[1] + DT_ * nv1;
    px[3*p+2] = Xv[2] + DT_ * nv2;
    pv[3*p] = nv0; pv[3*p+1] = nv1; pv[3*p+2] = nv2;
    #pragma unroll
    for (int i = 0; i < 9; ++i) pC[9*p+i] = C_COEF_ * Cn[i];
}

// ---------------------------------------------------------------- per-frame output
// d_out[frame] = grid_m + v. Stage grid_m through LDS via the Tensor Data Mover
// (tensor_load_to_lds, TENSORcnt). The bias-add runs on the matrix pipe:
// D = A x B + C with C = staged tile and A x B a rank-1 broadcast of v
// (A col K=0 = v, B row K=0 = 1 => (AxB)[m][n] = v exactly, in f32).
// Since we add a uniform scalar, the WMMA C/D lane<->element permutation is
// a bijection and thus irrelevant: load 8 floats/lane, wmma, store 8 floats.
__global__ void __launch_bounds__(256)
frame_out(const float* __restrict__ gm, const float* __restrict__ vptr,
          float* __restrict__ out, int total) {
    __shared__ __align__(16) float tile[CHUNK];
    const int start = blockIdx.x * CHUNK;

    if (threadIdx.x < 32) {   // wave 0 issues the TDM descriptor (EXEC ignored by TDM)
        unsigned long long gaddr = (unsigned long long)(uintptr_t)(gm + start);
        unsigned ldsa = (unsigned)(uintptr_t)(void*)tile;   // flat LDS addr: low 32 bits = LDS offset
        unsigned rem  = (unsigned)(total - start);          // tensor_dim0: reads at X>=rem return 0

        v4u g0;
        g0[0] = 1u;                                         // count=1, user descriptor
        g0[1] = ldsa;                                       // lds_addr
        g0[2] = (unsigned)(gaddr & 0xffffffffull);          // global_addr[31:0]
        g0[3] = (unsigned)((gaddr >> 32) & 0x1ffffffull)    // global_addr[56:32]
              | (2u << 30);                                 // type = 2 ("image")
        v8i_ g1;
        g1[0] = (int)(2u << 16);                            // data_size = 2 -> 4 bytes
        g1[1] = (int)((rem & 0xffffu) << 16);               // tensor_dim0[15:0]
        g1[2] = (int)((rem >> 16) & 0xffffu);               // tensor_dim0[31:16] | tensor_dim1=0
        g1[3] = (int)((unsigned)CHUNK << 16);               // tile_dim0 = CHUNK
        g1[4] = 0;                                          // tile_dim1=0, tile_dim2=0 (unused)
        g1[5] = (int)rem;                                   // tensor_dim0_stride[31:0]
        g1[6] = 0;
        g1[7] = 0;
        v4i_ g2 = {0, 0, 0, 0};
        v4i_ g3 = {0, 0, 0, 0};
#if __has_include(<hip/amd_detail/amd_gfx1250_TDM.h>)
        v8i_ gz = {0, 0, 0, 0, 0, 0, 0, 0};
        __builtin_amdgcn_tensor_load_to_lds(g0, g1, g2, g3, gz, 0);   // clang-23 6-arg form
#else
        __builtin_amdgcn_tensor_load_to_lds(g0, g1, g2, g3, 0);       // ROCm 7.2 5-arg form
#endif
        __builtin_amdgcn_s_wait_tensorcnt(0);
    }
    __syncthreads();

    const int lane = threadIdx.x & 31;
    const int wave = threadIdx.x >> 5;
    const float v  = vptr[0];

    // Rank-1 operands for the broadcast: A(16x4): col K=0 = v; B(4x16): row K=0 = 1.
    // 32-bit A layout: lanes 0-15 hold K=0,1 in VGPR0,1; lanes 16-31 hold K=2,3.
    v2f a, b;
    a.x = (lane < 16) ? v    : 0.0f;  a.y = 0.0f;
    b.x = (lane < 16) ? 1.0f : 0.0f;  b.y = 0.0f;

    for (int t = wave; t < CHUNK / 256; t += 8) {   // 256 floats per wave-tile
        const int off = t * 256 + lane;
        v8f c;
        #pragma unroll
        for (int r = 0; r < 8; ++r) c[r] = tile[off + 32 * r];
        v8f d = __builtin_amdgcn_wmma_f32_16x16x4_f32(
            /*neg_a=*/false, a, /*neg_b=*/false, b,
            /*c_mod=*/(short)0, c, /*reuse_a=*/false, /*reuse_b=*/false);
        #pragma unroll
        for (int r = 0; r < 8; ++r) {
            int gidx = start + off + 32 * r;
            if (gidx < total) out[gidx] = d[r];
        }
    }
}

// ---------------------------------------------------------------- driver
extern "C" void kernel_launch(void* const* d_in, const int* in_sizes, int n_in,
                              void* d_out, int out_size, void* d_ws, size_t ws_size,
                              hipStream_t stream) {
    const float* vptr = (const float*)d_in[0];
    const float* q    = (const float*)d_in[1];
    const float* qd   = (const float*)d_in[2];
    const int N = in_sizes[1] / 3;          // 13500 particles

    float* ws  = (float*)d_ws;
    float* gm  = ws;                        // G3
    float* gvx = ws + (size_t)G3;           // G3
    float* gvy = ws + (size_t)2 * G3;
    float* gvz = ws + (size_t)3 * G3;
    float* px  = ws + (size_t)4 * G3;       // 3N
    float* pv  = px + 3 * (size_t)N;        // 3N
    float* pC  = pv + 3 * (size_t)N;        // 9N
    float* pF  = pC + 9 * (size_t)N;        // 9N
    float* out = (float*)d_out;

    init_particles<<<(N + 255) / 256, 256, 0, stream>>>(q, qd, px, pv, pC, pF, N);

    const int pblocks = (N + 127) / 128;
    const int cblocks = (G3 + 255) / 256;
    const int oblocks = (G3 + CHUNK - 1) / CHUNK;

    for (int f = 0; f < 20; ++f) {
        for (int s = 0; s < 30; ++s) {
            clear_grid<<<cblocks, 256, 0, stream>>>((float4*)ws);
            p2g_kernel<<<pblocks, 128, 0, stream>>>(px, pv, pC, pF, gvx, gvy, gvz, gm, N);
            g2p_kernel<<<pblocks, 128, 0, stream>>>(px, pv, pC, gvx, gvy, gvz, gm, vptr, N);
        }
        frame_out<<<oblocks, 256, 0, stream>>>(gm, vptr, out + (size_t)f * G3, G3);
    }
}